// TAGAT_43602507989158
// MI455X (gfx1250) — compile-verified
//
#include <hip/hip_runtime.h>

typedef __attribute__((ext_vector_type(2))) float v2f;
typedef __attribute__((ext_vector_type(8))) float v8f;

#define Bn   8
#define Cn   1024
#define DIN  128
#define DHID 256
#define TENC 32
#define Hn   4
#define DK   64

// f32 WMMA: D(16x16,f32) = A(16x4,f32) x B(4x16,f32) + C
__device__ __forceinline__ v8f wmma4(v2f a, v2f b, v8f c) {
  return __builtin_amdgcn_wmma_f32_16x16x4_f32(
      /*neg_a=*/false, a, /*neg_b=*/false, b,
      /*c_mod=*/(short)0, c, /*reuse_a=*/false, /*reuse_b=*/false);
}

// ---------------------------------------------------------------------------
// 1) x_proj = x @ Wp^T + bp   (M=8192, N=256, K=128), one wave per 16x16 tile
// ---------------------------------------------------------------------------
__global__ __launch_bounds__(256) void proj_kernel(
    const float* __restrict__ x, const float* __restrict__ Wp,
    const float* __restrict__ bp, float* __restrict__ xp) {
  int wid  = blockIdx.x * (blockDim.x >> 5) + (threadIdx.x >> 5); // 0..8191
  int lane = threadIdx.x & 31;
  int mt = wid >> 4;          // 0..511
  int nt = wid & 15;          // 0..15
  int m0 = mt << 4, n0 = nt << 4;
  int rl = lane & 15, kh = lane >> 4;

  int row = m0 + rl;          // A row (M)
  int col = n0 + rl;          // B/D col (N)
  float bias = bp[col];
  v8f c;
#pragma unroll
  for (int r = 0; r < 8; ++r) c[r] = bias;

  const float* ax = x  + (size_t)row * DIN + 2 * kh;
  const float* bw = Wp + (size_t)col * DIN + 2 * kh;  // B[k][n] = Wp[n][k]
  for (int k0 = 0; k0 < DIN; k0 += 4) {
    v2f a, b;
    a.x = ax[k0]; a.y = ax[k0 + 1];
    b.x = bw[k0]; b.y = bw[k0 + 1];
    c = wmma4(a, b, c);
  }
#pragma unroll
  for (int r = 0; r < 8; ++r) {
    int orow = m0 + r + 8 * kh;
    xp[(size_t)orow * DHID + col] = c[r];
  }
}

// ---------------------------------------------------------------------------
// 2) s_src/s_dst, stored TRANSPOSED as (b,h,j) for stride-1 j-scans
// ---------------------------------------------------------------------------
__global__ __launch_bounds__(256) void sdot_kernel(
    const float* __restrict__ xp, const float* __restrict__ watt,
    float* __restrict__ ssrc, float* __restrict__ sdst) {
  int t = blockIdx.x * blockDim.x + threadIdx.x;   // (b*1024+j)*4+h
  if (t >= Bn * Cn * Hn) return;
  const float* vrow = xp + (size_t)t * DK;
  float as = 0.f, ad = 0.f;
#pragma unroll 8
  for (int d = 0; d < DK; ++d) {
    float xv = vrow[d];
    as += xv * watt[d];
    ad += xv * watt[DK + d];
  }
  int h = t & 3;
  int bj = t >> 2;
  int b = bj >> 10, j = bj & 1023;
  int o = (((b << 2) + h) << 10) + j;   // (b,h,j)
  ssrc[o] = as;
  sdst[o] = ad;
}

// ---------------------------------------------------------------------------
// 3) v[k] = sum_m w_t[m]*W2[m,k];  c0 = w_t . b2
// ---------------------------------------------------------------------------
__global__ void prep_kernel(const float* __restrict__ W2,
                            const float* __restrict__ b2,
                            const float* __restrict__ watt,
                            float* __restrict__ vbuf) {
  int k = threadIdx.x;
  if (k < TENC) {
    float acc = 0.f;
    for (int m = 0; m < TENC; ++m) acc += watt[2 * DK + m] * W2[m * TENC + k];
    vbuf[k] = acc;
  }
  if (k == TENC) {
    float acc = 0.f;
    for (int m = 0; m < TENC; ++m) acc += watt[2 * DK + m] * b2[m];
    vbuf[TENC] = acc;
  }
}

// ---------------------------------------------------------------------------
// 4) s_t[i,j] = c0 + sum_k v[k]*relu(W1[k]*(p_i - p_j) + b1[k])
// ---------------------------------------------------------------------------
__global__ __launch_bounds__(256) void st_kernel(
    const float* __restrict__ p, const float* __restrict__ W1,
    const float* __restrict__ b1, const float* __restrict__ vbuf,
    float* __restrict__ st) {
  int t = blockIdx.x * blockDim.x + threadIdx.x;   // i*1024+j
  int i = t >> 10, j = t & 1023;
  float tau = p[i] - p[j];
  float acc = vbuf[TENC];
#pragma unroll
  for (int k = 0; k < TENC; ++k) {
    float u = fmaf(W1[k], tau, b1[k]);
    u = fmaxf(u, 0.f);
    acc = fmaf(vbuf[k], u, acc);
  }
  st[t] = acc;
}

// ---------------------------------------------------------------------------
// 5) Attention: one wave per (b, 16-row i-tile, h).
//    Pass 1: pure max scan (float4 streams, no transcendentals).
//    Pass 2: un-normalized exp A-fragments x WMMA, sum accumulated inline;
//            per-row 1/S applied to C fragments at the end via shuffles.
// ---------------------------------------------------------------------------
__global__ __launch_bounds__(256) void attn_kernel(
    const float* __restrict__ xp, const float* __restrict__ ssrc,
    const float* __restrict__ sdst, const float* __restrict__ st,
    float* __restrict__ out) {
  int wid  = blockIdx.x * (blockDim.x >> 5) + (threadIdx.x >> 5); // 0..2047
  int lane = threadIdx.x & 31;
  int b   = wid >> 8;        // 64 i-tiles * 4 heads = 256 per batch
  int rem = wid & 255;
  int it  = rem >> 2, h = rem & 3;
  int i0  = it << 4;
  int rl  = lane & 15, kh = lane >> 4;
  int i   = i0 + rl;

  int bh = ((b << 2) + h) << 10;              // (b,h) plane, length 1024 in j
  float sd = sdst[bh + i];
  const float* srow  = ssrc + bh;             // stride-1 over j
  const float* strow = st + (size_t)i * Cn;   // stride-1 over j

  // ---- pass 1: row max; the two lane-halves split the j-range ----
  float m = -3.0e38f;
  int jbase = kh << 9;   // 0 or 512
  const float* s1 = srow  + jbase;
  const float* t1 = strow + jbase;
  for (int jj = 0; jj < 512; jj += 4) {
    float4 a4 = *(const float4*)(s1 + jj);
    float4 t4 = *(const float4*)(t1 + jj);
    float e0 = sd + a4.x + t4.x; e0 = (e0 > 0.f) ? e0 : 0.2f * e0;
    float e1 = sd + a4.y + t4.y; e1 = (e1 > 0.f) ? e1 : 0.2f * e1;
    float e2 = sd + a4.z + t4.z; e2 = (e2 > 0.f) ? e2 : 0.2f * e2;
    float e3 = sd + a4.w + t4.w; e3 = (e3 > 0.f) ? e3 : 0.2f * e3;
    m = fmaxf(m, fmaxf(fmaxf(e0, e1), fmaxf(e2, e3)));
  }
  float Mx = fmaxf(m, __shfl_xor(m, 16, 32));   // full-row max

  // ---- pass 2: exp tiles x x_proj via WMMA; accumulate row sum inline ----
  v8f c0v, c1v, c2v, c3v;
#pragma unroll
  for (int r = 0; r < 8; ++r) { c0v[r] = 0.f; c1v[r] = 0.f; c2v[r] = 0.f; c3v[r] = 0.f; }

  float s = 0.f;
  const float* xpb = xp + ((size_t)(b << 10)) * DHID + h * DK; // + j*DHID + d
  for (int j0 = 0; j0 < Cn; j0 += 4) {
    int ja = j0 + 2 * kh;
    float2 sv = *(const float2*)(srow + ja);
    float2 tv = *(const float2*)(strow + ja);
    float e0 = sd + sv.x + tv.x; e0 = (e0 > 0.f) ? e0 : 0.2f * e0;
    float e1 = sd + sv.y + tv.y; e1 = (e1 > 0.f) ? e1 : 0.2f * e1;
    float p0 = __expf(e0 - Mx);
    float p1 = __expf(e1 - Mx);
    s += p0 + p1;
    v2f a; a.x = p0; a.y = p1;

    const float* brow0 = xpb + (size_t)ja * DHID;
    const float* brow1 = brow0 + DHID;
    v2f bb;
    bb.x = brow0[rl];      bb.y = brow1[rl];      c0v = wmma4(a, bb, c0v);
    bb.x = brow0[16 + rl]; bb.y = brow1[16 + rl]; c1v = wmma4(a, bb, c1v);
    bb.x = brow0[32 + rl]; bb.y = brow1[32 + rl]; c2v = wmma4(a, bb, c2v);
    bb.x = brow0[48 + rl]; bb.y = brow1[48 + rl]; c3v = wmma4(a, bb, c3v);
  }

  // full-row sum for row rl (halves cover complementary j mod-4 pairs)
  float S   = s + __shfl_xor(s, 16, 32);
  float inv = 1.0f / S;

  // normalize C fragments: row of VGPR r is (r + 8*kh); fetch its 1/S
  float* obase = out + ((size_t)((b << 10) + i0)) * DHID + h * DK + rl;
#pragma unroll
  for (int r = 0; r < 8; ++r) {
    float invr = __shfl(inv, r + 8 * kh, 32);
    float* op = obase + (size_t)(r + 8 * kh) * DHID;
    op[0]  = c0v[r] * invr;
    op[16] = c1v[r] * invr;
    op[32] = c2v[r] * invr;
    op[48] = c3v[r] * invr;
  }
}

// ---------------------------------------------------------------------------
extern "C" void kernel_launch(void* const* d_in, const int* in_sizes, int n_in,
                              void* d_out, int out_size, void* d_ws, size_t ws_size,
                              hipStream_t stream) {
  const float* x    = (const float*)d_in[0];
  const float* Wp   = (const float*)d_in[1];
  const float* bp   = (const float*)d_in[2];
  const float* p    = (const float*)d_in[3];
  const float* W1   = (const float*)d_in[4];
  const float* b1   = (const float*)d_in[5];
  const float* W2   = (const float*)d_in[6];
  const float* b2   = (const float*)d_in[7];
  const float* watt = (const float*)d_in[8];
  float* out = (float*)d_out;

  float* ws   = (float*)d_ws;
  float* xp   = ws;                 // 8192*256        = 2,097,152 f
  float* ssrc = ws + 2097152;       // 8*4*1024 (b,h,j)=    32,768 f
  float* sdst = ws + 2129920;       //                 =    32,768 f
  float* vbuf = ws + 2162688;       // 33 (pad to 64)
  float* st   = ws + 2162752;       // 1024*1024       = 1,048,576 f
  (void)in_sizes; (void)n_in; (void)out_size; (void)ws_size;

  hipLaunchKernelGGL(proj_kernel, dim3(1024), dim3(256), 0, stream, x, Wp, bp, xp);
  hipLaunchKernelGGL(sdot_kernel, dim3(128),  dim3(256), 0, stream, xp, watt, ssrc, sdst);
  hipLaunchKernelGGL(prep_kernel, dim3(1),    dim3(64),  0, stream, W2, b2, watt, vbuf);
  hipLaunchKernelGGL(st_kernel,   dim3(4096), dim3(256), 0, stream, p, W1, b1, vbuf, st);
  hipLaunchKernelGGL(attn_kernel, dim3(256),  dim3(256), 0, stream, xp, ssrc, sdst, st, out);
}